// Communication_21337397526886
// MI455X (gfx1250) — compile-verified
//
#include <hip/hip_runtime.h>
#include <hip/hip_bf16.h>

#define NBOX 20
#define KPAD 20          // K stride in interleaved tables (floats); even pairs stay 8B aligned
#define CCH  256
#define HH   512
#define WW   512

typedef __attribute__((ext_vector_type(2))) float v2f;
typedef __attribute__((ext_vector_type(8))) float v8f;

// ---------------------------------------------------------------------------
// Kernel 1: box reduction + bilinear feature sampling + separable gaussian
// tables, stored K-innermost so the GEMM can load (k, k+1) pairs as b64:
//   ws_feats[c*20 + n] = bilinear(feat[c], cy_n, cx_n) / 20
//   ws_gy [h*20 + n]   = exp(-(h - x1_n)^2 / (2 bev_n^2))
//   ws_gx [w*20 + n]   = exp(-(w)^2        / (2 bev_n^2))
// ---------------------------------------------------------------------------
__global__ __launch_bounds__(256)
void comm_prep_kernel(const float* __restrict__ boxes,
                      const float* __restrict__ feat,
                      float* __restrict__ ws_feats,
                      float* __restrict__ ws_gy,
                      float* __restrict__ ws_gx)
{
    __shared__ float s_x1[NBOX], s_inv[NBOX], s_ly[NBOX], s_lx[NBOX];
    __shared__ int   s_yl[NBOX], s_xl[NBOX], s_yh[NBOX], s_xh[NBOX];

    const int t = threadIdx.x;
    if (t < NBOX) {
        float mnx = 1e30f, mny = 1e30f, mxx = -1e30f, mxy = -1e30f;
        #pragma unroll
        for (int i = 0; i < 8; ++i) {
            float x = boxes[(t * 8 + i) * 3 + 0];
            float y = boxes[(t * 8 + i) * 3 + 1];
            mnx = fminf(mnx, x); mxx = fmaxf(mxx, x);
            mny = fminf(mny, y); mxy = fmaxf(mxy, y);
        }
        const float inv_vox = 1.0f / 0.4f;           // 1/(VOXEL*DOWNSAMPLE)
        float x1 = (mnx + 102.4f) * inv_vox;
        float y1 = (mny + 102.4f) * inv_vox;
        float x2 = (mxx + 102.4f) * inv_vox;
        float y2 = (mxy + 102.4f) * inv_vox;
        float bev = (y2 - y1) * (x2 - x1);
        float cx = 0.5f * (x1 + x2);
        float cy = 0.5f * (y1 + y2);

        s_x1[t]  = x1;
        s_inv[t] = 1.0f / (2.0f * bev * bev);

        float y = fminf(fmaxf(cy, 0.0f), (float)(HH - 1));
        float x = fminf(fmaxf(cx, 0.0f), (float)(WW - 1));
        int yl = (int)floorf(y);
        int xl = (int)floorf(x);
        s_yl[t] = yl; s_xl[t] = xl;
        s_yh[t] = min(yl + 1, HH - 1);
        s_xh[t] = min(xl + 1, WW - 1);
        s_ly[t] = y - (float)yl;
        s_lx[t] = x - (float)xl;
    }
    __syncthreads();

    // Bilinear-sampled per-box channel features, 1/NBOX folded in. K-innermost.
    for (int idx = t; idx < NBOX * CCH; idx += 256) {
        int n = idx >> 8;
        int c = idx & 255;
        int yl = s_yl[n], xl = s_xl[n], yh = s_yh[n], xh = s_xh[n];
        float ly = s_ly[n], lx = s_lx[n];
        const float* fc = feat + (size_t)c * HH * WW;
        float v00 = fc[yl * WW + xl];
        float v01 = fc[yl * WW + xh];
        float v10 = fc[yh * WW + xl];
        float v11 = fc[yh * WW + xh];
        float v = v00 * (1.0f - ly) * (1.0f - lx) + v01 * (1.0f - ly) * lx
                + v10 * ly * (1.0f - lx)          + v11 * ly * lx;
        ws_feats[c * KPAD + n] = v * (1.0f / NBOX);
    }

    // Separable gaussian factors, K-innermost (gy over rows h, gx over cols w).
    for (int idx = t; idx < NBOX * HH; idx += 256) {
        int n = idx >> 9;
        int h = idx & 511;
        float inv = s_inv[n];
        float dy = (float)h - s_x1[n];
        ws_gy[h * KPAD + n] = __expf(-dy * dy * inv);
        float w = (float)h;
        ws_gx[h * KPAD + n] = __expf(-w * w * inv);
    }
}

// ---------------------------------------------------------------------------
// Kernel 2: out[(c,h), w] = sum_k (feats[k,c]*gy[k,h]) * gx[k,w]
// One wave per group of 4 consecutive 16x16 w-tiles (A fragment reused).
// K = 20 -> 5 x V_WMMA_F32_16X16X4_F32 per tile.
// A layout (16x4 f32): lanes 0-15 hold {K=0,K=1}, lanes 16-31 hold {K=2,K=3},
// M = lane&15.  B layout (4x16 f32): mirrored, N = lane&15.
// With K-interleaved tables, each (k0,k0+1) pair is one aligned b64 load
// (k0 = 4*kb + khalf is always even).
// D layout (16x16 f32): VGPR i -> row i (lanes 0-15) / row i+8 (lanes 16-31).
// Output stores are non-temporal: 256 MB write-once stream, keep L2 for the
// hot 100 KB gy/gx/feats tables.
// ---------------------------------------------------------------------------
__global__ __launch_bounds__(256)
void comm_gemm_kernel(const float* __restrict__ ws_feats,
                      const float* __restrict__ ws_gy,
                      const float* __restrict__ ws_gx,
                      float* __restrict__ out)
{
    const int lane  = threadIdx.x & 31;
    const int wave  = threadIdx.x >> 5;
    const int TI    = blockIdx.x * 8 + wave;      // 0 .. 65535 tile-groups

    const int twg   = TI & 7;                     // w-tile-group (4 tiles each)
    const int th    = (TI >> 3) & 31;             // h tile
    const int c     = TI >> 8;                    // channel 0..255

    const int h0    = th * 16;
    const int wbase = twg * 64;

    const int row   = lane & 15;                  // M (for A) / N (for B)
    const int khalf = (lane >> 4) << 1;           // 0 or 2

    // Build the A fragment once: A[m, k] = feats[k,c] * gy[k, h0+m].
    // Each kb needs k = {k0, k0+1}: one b64 load from feats, one from gy.
    v2f afrag[5];
    #pragma unroll
    for (int kb = 0; kb < 5; ++kb) {
        int k0 = kb * 4 + khalf;                  // even -> 8B aligned pairs
        v2f f = *(const v2f*)(ws_feats + c * KPAD + k0);
        v2f g = *(const v2f*)(ws_gy + (h0 + row) * KPAD + k0);
        afrag[kb].x = f.x * g.x;
        afrag[kb].y = f.y * g.y;
    }

    #pragma unroll
    for (int s = 0; s < 4; ++s) {
        const int w0 = wbase + s * 16;
        v8f acc = {};
        #pragma unroll
        for (int kb = 0; kb < 5; ++kb) {
            int k0 = kb * 4 + khalf;
            v2f b = *(const v2f*)(ws_gx + (w0 + row) * KPAD + k0);
            acc = __builtin_amdgcn_wmma_f32_16x16x4_f32(
                /*neg_a=*/false, afrag[kb], /*neg_b=*/false, b,
                /*c_mod=*/(short)0, acc, /*reuse_a=*/false, /*reuse_b=*/false);
        }
        // Store 16x16 f32 tile per the C/D VGPR layout (streaming / NT).
        size_t base = ((size_t)c * HH + (size_t)(h0 + ((lane >> 4) << 3))) * WW
                    + (size_t)(w0 + row);
        #pragma unroll
        for (int i = 0; i < 8; ++i) {
            __builtin_nontemporal_store(acc[i], &out[base + (size_t)i * WW]);
        }
    }
}

extern "C" void kernel_launch(void* const* d_in, const int* in_sizes, int n_in,
                              void* d_out, int out_size, void* d_ws, size_t ws_size,
                              hipStream_t stream) {
    const float* boxes = (const float*)d_in[0];   // (20, 8, 3) f32
    const float* feat  = (const float*)d_in[1];   // (1, 256, 512, 512) f32
    float* out = (float*)d_out;                   // (1, 256, 512, 512) f32

    float* ws       = (float*)d_ws;
    float* ws_feats = ws;                         // 256*20 = 5120 floats
    float* ws_gy    = ws + CCH * KPAD;            // 512*20 = 10240 floats
    float* ws_gx    = ws_gy + HH * KPAD;          // 512*20 = 10240 floats

    comm_prep_kernel<<<1, 256, 0, stream>>>(boxes, feat, ws_feats, ws_gy, ws_gx);

    // 256 (c) * 32 (h tiles) * 8 (groups of 4 w-tiles) = 65536 wave jobs,
    // 8 waves / block -> 8192 blocks.
    comm_gemm_kernel<<<8192, 256, 0, stream>>>(ws_feats, ws_gy, ws_gx, out);
}